// UpSamplingWithIndices_38457137168658
// MI455X (gfx1250) — compile-verified
//
#include <hip/hip_runtime.h>

// Max-unpooling scatter-add, MI455X (gfx1250).
//
// Shapes fixed by the reference: x/argmax = [8,128,128,64] f32/i32,
// out = [8,256,256,64] f32.  Pure memory op: ~192 MiB of unavoidable HBM
// traffic -> ~8-10 us at 23.3 TB/s.  No matmul structure, so no WMMA; the
// CDNA5-specific paths used here are 128-bit vmem with TH=NT on the
// streamed inputs, RT zero-fill stores (keep the 128 MiB output resident
// in the 192 MB L2 so atomics are L2-local), and hardware
// global_atomic_add_f32 at DEV scope (no-return form).

typedef float v4f __attribute__((ext_vector_type(4)));
typedef int   v4i __attribute__((ext_vector_type(4)));

constexpr long long B_  = 8;
constexpr long long H_  = 128;
constexpr long long W_  = 128;
constexpr long long C_  = 64;
constexpr long long IMG  = H_ * W_ * C_;     // 2^20 pooled elems per image
constexpr long long OIMG = 4ll * IMG;        // 2^22 output elems per image
constexpr long long NIN  = B_ * IMG;         // 2^23 = 8,388,608
constexpr long long NOUT = B_ * OIMG;        // 2^25 = 33,554,432

// ---------------------------------------------------------------------------
// Kernel 1: zero the output with 128-bit regular-temporal stores.
// RT (not NT) on purpose: the zeroed lines must stay in L2 so the scatter
// atomics hit there instead of round-tripping HBM.
// ---------------------------------------------------------------------------
__global__ void unpool_zero_kernel(float* __restrict__ out) {
  long long i = ((long long)blockIdx.x * blockDim.x + threadIdx.x) * 4;
  v4f z = (v4f)0.0f;
  *(v4f*)(out + i) = z;   // global_store_b128, TH=RT
}

// ---------------------------------------------------------------------------
// Kernel 2: scatter-add.  4 elements per thread via B128 NT loads, then 4
// no-return DEV-scope f32 atomic adds.  flat = b*OIMG + (a & (OIMG-1)) is
// bit-exact with the reference's div/mod chain for these power-of-2 dims.
// ---------------------------------------------------------------------------
__global__ void unpool_scatter_kernel(const float* __restrict__ x,
                                      const int*  __restrict__ idx,
                                      float* __restrict__ out) {
  long long t    = (long long)blockIdx.x * blockDim.x + threadIdx.x;
  long long base = t * 4;

  // Streaming inputs: read-once, keep them out of L2's way.
  v4f v = __builtin_nontemporal_load((const v4f*)(x   + base));
  v4i a = __builtin_nontemporal_load((const v4i*)(idx + base));

  // All 4 consecutive pooled elements share one batch index (C % 4 == 0).
  long long obase = (base >> 20) << 22;   // (base / IMG) * OIMG

  const unsigned mask = (unsigned)(OIMG - 1);
#pragma unroll
  for (int j = 0; j < 4; ++j) {
    long long flat = obase + (long long)((unsigned)a[j] & mask);
    float* p = out + flat;
    // Hardware f32 atomic add, no return (STOREcnt path), device scope so the
    // RMW resolves at L2 (WGP scope would be incoherent across WGPs).
    asm volatile("global_atomic_add_f32 %0, %1, off scope:SCOPE_DEV"
                 :: "v"(p), "v"(v[j])
                 : "memory");
  }
}

// ---------------------------------------------------------------------------
extern "C" void kernel_launch(void* const* d_in, const int* in_sizes, int n_in,
                              void* d_out, int out_size, void* d_ws, size_t ws_size,
                              hipStream_t stream) {
  const float* x   = (const float*)d_in[0];
  const int*   idx = (const int*)d_in[1];
  float*       out = (float*)d_out;

  // Zero-fill: 2^25 floats / 4 per thread / 256 per block = 32768 blocks.
  {
    dim3 block(256);
    dim3 grid((unsigned)(NOUT / 4 / 256));
    unpool_zero_kernel<<<grid, block, 0, stream>>>(out);
  }

  // Scatter: 2^23 pooled elems / 4 per thread / 256 per block = 8192 blocks.
  // Same stream -> ordered after the zero-fill.
  {
    dim3 block(256);
    dim3 grid((unsigned)(NIN / 4 / 256));
    unpool_scatter_kernel<<<grid, block, 0, stream>>>(x, idx, out);
  }
}